// VanillaRNNLM_38809324486805
// MI455X (gfx1250) — compile-verified
//
#include <hip/hip_runtime.h>
#include <hip/hip_bf16.h>

// ---------------------------------------------------------------------------
// VanillaRNNLM on MI455X (gfx1250): f32 data -> V_WMMA_F32_16X16X4_F32,
// global->LDS staging via GLOBAL_LOAD_ASYNC_TO_LDS_B128 (ASYNCcnt) with a
// double-buffered K pipeline.  128x128 block tile, 8 waves in 4(M)x2(N),
// each wave computes a 32x64 tile (2x4 subtiles of 16x16).
//   phase 1: gather x = emb[tokens] (pad row 0 zeroed)
//   phase 2: xproj = x @ Wxh + bias            (WMMA GEMM, 4096x1024x512)
//   phase 3: 256 sequential steps h = tanh(xp_t + h @ Whh)   (WMMA, M=16)
//   phase 4: logits = hs @ Wout + bout         (WMMA GEMM, 4096x32000x1024)
// ---------------------------------------------------------------------------

typedef float v2f __attribute__((ext_vector_type(2)));
typedef float v4f __attribute__((ext_vector_type(4)));
typedef float v8f __attribute__((ext_vector_type(8)));
typedef int   v4i_t __attribute__((vector_size(16)));   // matches builtin proto

#define B_SZ   16
#define T_SZ   256
#define EMBED  512
#define HIDDEN 1024
#define VOCAB  32000
#define ROWS   (B_SZ * T_SZ)        // 4096

// ---- gfx1250 async global->LDS path (guarded; synchronous fallback) -------
#if __has_builtin(__builtin_amdgcn_global_load_async_to_lds_b128)
#define HAVE_ASYNC_LDS 1
__device__ __forceinline__ void async_copy16(const float* g, float* l) {
    __builtin_amdgcn_global_load_async_to_lds_b128(
        (__attribute__((address_space(1))) v4i_t*)(float*)(g),
        (__attribute__((address_space(3))) v4i_t*)(l), 0, 0);
}
#if __has_builtin(__builtin_amdgcn_s_wait_asynccnt)
#define WAIT_ASYNC(n) __builtin_amdgcn_s_wait_asynccnt(n)
#else
#define WAIT_ASYNC(n) asm volatile("s_wait_asynccnt %0" ::"i"(n) : "memory")
#endif
#else
__device__ __forceinline__ void async_copy16(const float* g, float* l) {
    *(v4f*)(l) = *(const v4f*)(g);
}
#define WAIT_ASYNC(n) ((void)0)
#endif

// ---------------------------------------------------------------------------
// Phase 1: embedding gather with padding_idx = 0
// ---------------------------------------------------------------------------
__global__ __launch_bounds__(256) void gather_embed_kernel(
    const int* __restrict__ tokens,   // [4096]
    const float* __restrict__ emb,    // [VOCAB][EMBED]
    float* __restrict__ x)            // [4096][EMBED]
{
    const int i = blockIdx.x * 256 + threadIdx.x;      // 0 .. 4096*512-1
    const int row = i >> 9;                            // / EMBED
    const int e   = i & (EMBED - 1);
    const int tok = tokens[row];
    x[i] = (tok == 0) ? 0.0f : emb[(size_t)tok * EMBED + e];
}

// ---------------------------------------------------------------------------
// Generic f32 WMMA GEMM:  C[M][N] = A[M][K] @ B[K][N] + bias[N]
// Block tile 128(M) x 128(N), K-chunk 32, 8 waves (wave32) per block in a
// 4(M) x 2(N) grid; each wave owns 32 rows x 64 cols = 2x4 16x16 subtiles.
// Double-buffered LDS, tiles staged with async global->LDS b128 copies.
//
// Fragment layouts (ISA 7.12.2, 32-bit, wave32):
//   A 16x4 : lane = M (lane%16); VGPR0 = K + (lane<16?0:2), VGPR1 = +1
//   B 4x16 : lane = N (lane%16); VGPR0 = K + (lane<16?0:2), VGPR1 = +1
//   C 16x16: lane = N (lane%16); VGPR r = M = r + 8*(lane>=16)
// As row-major [m][k], row stride 36 (16B-aligned rows; half-wave K+2 offset
// lands on disjoint bank sets).  Bs row-major [k][n], row stride 136
// (2*136 mod 64 = 16 -> halves on disjoint banks).
// ---------------------------------------------------------------------------
#define GBM 128
#define GBN 128
#define GBK 32
#define A_LD 36
#define B_LD 136
#define ASYNC_PER_THREAD 8          // 4 for A tile + 4 for B tile

__device__ __forceinline__ void stage_tiles(
    const float* __restrict__ A, const float* __restrict__ B,
    int K, int N, int m0, int n0, int k0,
    float* Asb, float* Bsb, int tid)
{
    // A tile: 128 x 32 floats, one b128 per lane x4
    #pragma unroll
    for (int p = 0; p < 4; ++p) {
        const int row = (tid >> 3) + p * 32;        // 0..127
        const int kq  = (tid & 7) * 4;              // 0,4,..,28
        async_copy16(&A[(size_t)(m0 + row) * K + k0 + kq],
                     &Asb[row * A_LD + kq]);
    }
    // B tile: 32 x 128 floats, one b128 per lane x4
    #pragma unroll
    for (int p = 0; p < 4; ++p) {
        const int row = (tid >> 5) + p * 8;         // 0..31
        const int nq  = (tid & 31) * 4;             // 0..124
        async_copy16(&B[(size_t)(k0 + row) * N + n0 + nq],
                     &Bsb[row * B_LD + nq]);
    }
}

__global__ __launch_bounds__(256) void wmma_gemm_f32_kernel(
    const float* __restrict__ A, const float* __restrict__ B,
    const float* __restrict__ bias, float* __restrict__ C,
    int M, int N, int K)
{
    __shared__ float As[2][GBM * A_LD];
    __shared__ float Bs[2][GBK * B_LD];

    const int tid  = threadIdx.x;
    const int wave = tid >> 5;
    const int lane = tid & 31;
    const int half = lane >> 4;     // 0: K=0,1   1: K=2,3
    const int l16  = lane & 15;
    const int m0   = blockIdx.x * GBM;
    const int n0   = blockIdx.y * GBN;
    const int wm   = (wave & 3) * 32;   // wave's M offset inside block tile
    const int wn   = (wave >> 2) * 64;  // wave's N offset inside block tile

    v8f acc[2][4] = {};

    const int nch = K / GBK;
    stage_tiles(A, B, K, N, m0, n0, 0, As[0], Bs[0], tid);

    for (int ic = 0; ic < nch; ++ic) {
        const int cur = ic & 1;
        if (ic + 1 < nch) {
            stage_tiles(A, B, K, N, m0, n0, (ic + 1) * GBK,
                        As[cur ^ 1], Bs[cur ^ 1], tid);
            WAIT_ASYNC(ASYNC_PER_THREAD);   // prior chunk's copies retired
        } else {
            WAIT_ASYNC(0);
        }
        __syncthreads();

        const float* Ab = As[cur];
        const float* Bb = Bs[cur];
        const int arow0 = (wm + l16) * A_LD + 2 * half;

        #pragma unroll
        for (int kk = 0; kk < GBK; kk += 4) {
            v2f a[2];                               // adjacent -> ds_load_b64
            #pragma unroll
            for (int sm = 0; sm < 2; ++sm) {
                a[sm].x = Ab[arow0 + sm * 16 * A_LD + kk + 0];
                a[sm].y = Ab[arow0 + sm * 16 * A_LD + kk + 1];
            }
            #pragma unroll
            for (int s = 0; s < 4; ++s) {
                const int bcol = wn + s * 16 + l16;
                v2f b;
                b.x = Bb[(kk + 2 * half + 0) * B_LD + bcol];
                b.y = Bb[(kk + 2 * half + 1) * B_LD + bcol];
                acc[0][s] = __builtin_amdgcn_wmma_f32_16x16x4_f32(
                    false, a[0], false, b, (short)0, acc[0][s], false, false);
                acc[1][s] = __builtin_amdgcn_wmma_f32_16x16x4_f32(
                    false, a[1], false, b, (short)0, acc[1][s], false, false);
            }
        }
        __syncthreads();
    }

    // --- epilogue: bias add + store ---
    #pragma unroll
    for (int sm = 0; sm < 2; ++sm) {
        const int row_base = m0 + wm + sm * 16 + half * 8;
        #pragma unroll
        for (int s = 0; s < 4; ++s) {
            const int col = n0 + wn + s * 16 + l16;
            const float bv = bias[col];
            #pragma unroll
            for (int r = 0; r < 8; ++r) {
                C[(size_t)(row_base + r) * N + col] = acc[sm][s][r] + bv;
            }
        }
    }
}

// ---------------------------------------------------------------------------
// Phase 3: one recurrence step  hs[:,t,:] = tanh(xproj[:,t,:] + h_prev @ Whh)
// h_prev = hs[:,t-1,:] (zeros for t==0).  M = 16 (batch) -> one WMMA M-tile.
// Grid: 4 blocks x 256 threads; each wave owns 32 columns (2 subtiles).
// h_prev staged transposed in LDS hpT[k][b]: bank = (16k+b)%64; half-wave
// K-offset (+2) shifts banks by 32 -> conflict-free.
// ---------------------------------------------------------------------------
__global__ __launch_bounds__(256) void rnn_step_kernel(
    const float* __restrict__ xproj,   // [4096][HIDDEN], row = b*T + t
    const float* __restrict__ Whh,     // [HIDDEN][HIDDEN]
    float* __restrict__ hs,            // [4096][HIDDEN]
    int t)
{
    __shared__ float hpT[HIDDEN * B_SZ];   // hpT[k*16 + b]

    const int tid = threadIdx.x;
    if (t == 0) {
        for (int i = tid; i < HIDDEN * B_SZ; i += 256) hpT[i] = 0.0f;
    } else {
        const int b   = tid >> 4;           // 0..15
        const int seg = tid & 15;           // 0..15
        const float* src = hs + (size_t)(b * T_SZ + (t - 1)) * HIDDEN;
        #pragma unroll 4
        for (int j = 0; j < 64; ++j) {
            const int k = seg * 64 + j;
            hpT[k * B_SZ + b] = src[k];
        }
    }
    __syncthreads();

    const int wave = tid >> 5;
    const int lane = tid & 31;
    const int half = lane >> 4;
    const int l16  = lane & 15;
    const int ncol0 = blockIdx.x * 256 + wave * 32;

    v8f acc[2] = {};
    for (int k0 = 0; k0 < HIDDEN; k0 += 4) {
        v2f a;
        a.x = hpT[(k0 + half * 2 + 0) * B_SZ + l16];
        a.y = hpT[(k0 + half * 2 + 1) * B_SZ + l16];
        #pragma unroll
        for (int s = 0; s < 2; ++s) {
            const int n = ncol0 + s * 16 + l16;
            v2f b;
            b.x = Whh[(size_t)(k0 + half * 2 + 0) * HIDDEN + n];
            b.y = Whh[(size_t)(k0 + half * 2 + 1) * HIDDEN + n];
            acc[s] = __builtin_amdgcn_wmma_f32_16x16x4_f32(
                false, a, false, b, (short)0, acc[s], false, false);
        }
    }

    #pragma unroll
    for (int s = 0; s < 2; ++s) {
        const int n = ncol0 + s * 16 + l16;
        #pragma unroll
        for (int r = 0; r < 8; ++r) {
            const int b = r + half * 8;                         // batch = M
            const size_t idx = (size_t)(b * T_SZ + t) * HIDDEN + n;
            hs[idx] = tanhf(xproj[idx] + acc[s][r]);
        }
    }
}

// ---------------------------------------------------------------------------
// launch
// ---------------------------------------------------------------------------
extern "C" void kernel_launch(void* const* d_in, const int* in_sizes, int n_in,
                              void* d_out, int out_size, void* d_ws, size_t ws_size,
                              hipStream_t stream) {
    const int*   tokens = (const int*)  d_in[0];   // [16,256]
    const float* emb    = (const float*)d_in[1];   // [32000,512]
    const float* Wxh    = (const float*)d_in[2];   // [512,1024]
    const float* Whh    = (const float*)d_in[3];   // [1024,1024]
    const float* bias   = (const float*)d_in[4];   // [1024]
    const float* Wout   = (const float*)d_in[5];   // [1024,32000]
    const float* bout   = (const float*)d_in[6];   // [32000]
    float*       out    = (float*)d_out;           // [16,256,32000]

    float* ws    = (float*)d_ws;
    float* x     = ws;                                       // 4096*512
    float* xproj = ws + (size_t)ROWS * EMBED;                // 4096*1024
    float* hs    = xproj + (size_t)ROWS * HIDDEN;            // 4096*1024

    // phase 1: embedding gather
    gather_embed_kernel<<<(ROWS * EMBED) / 256, 256, 0, stream>>>(tokens, emb, x);

    // phase 2: xproj = x @ Wxh + bias
    {
        dim3 grid(ROWS / GBM, HIDDEN / GBN);
        wmma_gemm_f32_kernel<<<grid, 256, 0, stream>>>(x, Wxh, bias, xproj,
                                                       ROWS, HIDDEN, EMBED);
    }

    // phase 3: sequential recurrence (graph-friendly fixed launch sequence)
    for (int t = 0; t < T_SZ; ++t) {
        rnn_step_kernel<<<HIDDEN / 256, 256, 0, stream>>>(xproj, Whh, hs, t);
    }

    // phase 4: logits = hs @ Wout + bout
    {
        dim3 grid(ROWS / GBM, VOCAB / GBN);
        wmma_gemm_f32_kernel<<<grid, 256, 0, stream>>>(hs, Wout, bout, out,
                                                       ROWS, VOCAB, HIDDEN);
    }
}